// TransformerBlock_83640193122339
// MI455X (gfx1250) — compile-verified
//
#include <hip/hip_runtime.h>

typedef __attribute__((ext_vector_type(2))) float v2f;
typedef __attribute__((ext_vector_type(8))) float v8f;

#define D       64
#define KNN     16
#define NPTS    8192
#define BATCH   4
#define TOTAL_PTS (BATCH * NPTS)

__device__ __forceinline__ v8f wmma_f32(v2f a, v2f b, v8f c) {
  // V_WMMA_F32_16X16X4_F32 : D = A(16x4) * B(4x16) + C, full fp32
  return __builtin_amdgcn_wmma_f32_16x16x4_f32(false, a, false, b, (short)0, c,
                                               false, false);
}

// Y(16x64) = X(16x64) @ W(64x64) + bias, optional ReLU.
// One wave32. X in LDS, W in LDS, Y any addr space (row stride 64 floats).
template <bool RELU>
__device__ __forceinline__ void mm16x64_64x64(const float* __restrict__ X,
                                              const float* __restrict__ W,
                                              const float* __restrict__ bias,
                                              float* __restrict__ Y, int lane) {
  const int m = lane & 15;   // A row / B,C,D column within tile
  const int g = lane >> 4;   // lane group
  v8f acc[4];
#pragma unroll
  for (int nt = 0; nt < 4; ++nt) {
    const float bv = bias[nt * 16 + m];
    v8f c;
#pragma unroll
    for (int r = 0; r < 8; ++r) c[r] = bv;
#pragma unroll
    for (int kk = 0; kk < 16; ++kk) {
      const int kb = kk * 4 + 2 * g;
      v2f a, b;
      a[0] = X[m * D + kb];
      a[1] = X[m * D + kb + 1];
      b[0] = W[kb * D + nt * 16 + m];
      b[1] = W[(kb + 1) * D + nt * 16 + m];
      c = wmma_f32(a, b, c);
    }
    acc[nt] = c;
  }
  // write back only after all reads of X are done (in-place safe)
#pragma unroll
  for (int nt = 0; nt < 4; ++nt) {
#pragma unroll
    for (int r = 0; r < 8; ++r) {
      float v = acc[nt][r];
      if (RELU) v = v > 0.f ? v : 0.f;
      Y[(r + 8 * g) * D + nt * 16 + m] = v;
    }
  }
}

// ---------------- Kernel 1: x = features@fc1+b ; q/k/v = x@{wq,wk,wv} -------
__global__ void __launch_bounds__(128) proj_kernel(
    const float* __restrict__ features, const float* __restrict__ fc1_w,
    const float* __restrict__ fc1_b, const float* __restrict__ wq,
    const float* __restrict__ wk, const float* __restrict__ wv,
    float* __restrict__ qo, float* __restrict__ ko, float* __restrict__ vo) {
  extern __shared__ float smem[];
  float* s_fc1 = smem;               // 4096
  float* s_wq  = s_fc1 + 4096;       // 4096
  float* s_wk  = s_wq + 4096;        // 4096
  float* s_wv  = s_wk + 4096;        // 4096
  float* s_b   = s_wv + 4096;        // 64
  float* s_z   = s_b + 64;           // 64 (zero bias)
  float* s_feat = s_z + 64;          // 4 * 1024
  float* s_x    = s_feat + 4 * 1024; // 4 * 1024

  const int tid = threadIdx.x;
  for (int i = tid; i < 4096; i += 128) {
    s_fc1[i] = fc1_w[i];
    s_wq[i]  = wq[i];
    s_wk[i]  = wk[i];
    s_wv[i]  = wv[i];
  }
  if (tid < 64) { s_b[tid] = fc1_b[tid]; s_z[tid] = 0.f; }
  __syncthreads();

  const int wid = tid >> 5, lane = tid & 31;
  const int tile = blockIdx.x * 4 + wid;   // 0..2047
  const int p0 = tile * 16;                // 16 contiguous points
  float* feat = s_feat + wid * 1024;
  float* xb   = s_x + wid * 1024;

  for (int i = lane; i < 1024; i += 32) feat[i] = features[(size_t)p0 * D + i];

  mm16x64_64x64<false>(feat, s_fc1, s_b, xb, lane);              // x
  mm16x64_64x64<false>(xb, s_wq, s_z, qo + (size_t)p0 * D, lane); // q
  mm16x64_64x64<false>(xb, s_wk, s_z, ko + (size_t)p0 * D, lane); // k (pre-gather)
  mm16x64_64x64<false>(xb, s_wv, s_z, vo + (size_t)p0 * D, lane); // v (pre-gather)
}

// ---------------- Kernel 2: per-point attention (one wave = one point) ------
__global__ void __launch_bounds__(128) point_attn_kernel(
    const float* __restrict__ xyz, const float* __restrict__ features,
    const int* __restrict__ knn, const float* __restrict__ d1_w,
    const float* __restrict__ d1_b, const float* __restrict__ d2_w,
    const float* __restrict__ d2_b, const float* __restrict__ g1_w,
    const float* __restrict__ g1_b, const float* __restrict__ g2_w,
    const float* __restrict__ g2_b, const float* __restrict__ fc2_w,
    const float* __restrict__ fc2_b, const float* __restrict__ qi,
    const float* __restrict__ ki, const float* __restrict__ vi,
    float* __restrict__ out_res, float* __restrict__ out_attn) {
  extern __shared__ float smem[];
  float* s_d2  = smem;            // 4096
  float* s_g1  = s_d2 + 4096;     // 4096
  float* s_g2  = s_g1 + 4096;     // 4096
  float* s_fc2 = s_g2 + 4096;     // 4096
  float* s_d1  = s_fc2 + 4096;    // 192
  float* s_d1b = s_d1 + 192;      // 64
  float* s_d2b = s_d1b + 64;      // 64
  float* s_g1b = s_d2b + 64;      // 64
  float* s_g2b = s_g1b + 64;      // 64
  float* s_fc2b = s_g2b + 64;     // 64
  float* s_pw  = s_fc2b + 64;     // 4 * 2192 per-wave scratch

  const int tid = threadIdx.x;
  for (int i = tid; i < 4096; i += 128) {
    s_d2[i] = d2_w[i]; s_g1[i] = g1_w[i]; s_g2[i] = g2_w[i]; s_fc2[i] = fc2_w[i];
  }
  for (int i = tid; i < 192; i += 128) s_d1[i] = d1_w[i];
  if (tid < 64) {
    s_d1b[tid] = d1_b[tid]; s_d2b[tid] = d2_b[tid]; s_g1b[tid] = g1_b[tid];
    s_g2b[tid] = g2_b[tid]; s_fc2b[tid] = fc2_b[tid];
  }
  __syncthreads();

  const int wid = tid >> 5, lane = tid & 31;
  float* bufA = s_pw + wid * 2192;  // 16x64 working tile
  float* pos  = bufA + 1024;        // 16x64 pos_enc
  float* qrow = pos + 1024;         // 64
  float* resr = qrow + 64;          // 64
  int* idxs   = (int*)(resr + 64);  // 16

  const int f0 = lane, f1 = lane + 32;
  const int gw = blockIdx.x * 4 + wid;  // global wave id, 0..4095

  for (int p = gw; p < TOTAL_PTS; p += 4096) {
    const int b = p / NPTS;
    if (lane < KNN) idxs[lane] = knn[(size_t)p * KNN + lane];
    qrow[f0] = qi[(size_t)p * D + f0];
    qrow[f1] = qi[(size_t)p * D + f1];
    const float cx = xyz[(size_t)p * 3 + 0];
    const float cy = xyz[(size_t)p * 3 + 1];
    const float cz = xyz[(size_t)p * 3 + 2];

    // hidden = relu(rel @ d1 + d1_b)   -> bufA (16x64)
    for (int j = 0; j < KNN; ++j) {
      const size_t nb = (size_t)b * NPTS + idxs[j];
      const float rx = cx - xyz[nb * 3 + 0];
      const float ry = cy - xyz[nb * 3 + 1];
      const float rz = cz - xyz[nb * 3 + 2];
      float h0 = rx * s_d1[f0] + ry * s_d1[64 + f0] + rz * s_d1[128 + f0] + s_d1b[f0];
      float h1 = rx * s_d1[f1] + ry * s_d1[64 + f1] + rz * s_d1[128 + f1] + s_d1b[f1];
      bufA[j * D + f0] = h0 > 0.f ? h0 : 0.f;
      bufA[j * D + f1] = h1 > 0.f ? h1 : 0.f;
    }
    // pos_enc = hidden @ d2 + d2_b
    mm16x64_64x64<false>(bufA, s_d2, s_d2b, pos, lane);

    // h = q - k + pos_enc  -> bufA
    for (int j = 0; j < KNN; ++j) {
      const size_t nb = (size_t)b * NPTS + idxs[j];
      bufA[j * D + f0] = qrow[f0] - ki[nb * D + f0] + pos[j * D + f0];
      bufA[j * D + f1] = qrow[f1] - ki[nb * D + f1] + pos[j * D + f1];
    }
    // t = relu(h @ g1 + g1_b) ; attn_pre = t @ g2 + g2_b   (in place)
    mm16x64_64x64<true>(bufA, s_g1, s_g1b, bufA, lane);
    mm16x64_64x64<false>(bufA, s_g2, s_g2b, bufA, lane);

    // softmax over the 16 neighbors (axis=-2), per feature column
    float m0 = -1e30f, m1 = -1e30f;
    for (int j = 0; j < KNN; ++j) {
      m0 = fmaxf(m0, bufA[j * D + f0]);
      m1 = fmaxf(m1, bufA[j * D + f1]);
    }
    const float sc = 0.125f;  // 1/sqrt(64)
    float e0[KNN], e1[KNN], sum0 = 0.f, sum1 = 0.f;
    for (int j = 0; j < KNN; ++j) {
      e0[j] = __expf((bufA[j * D + f0] - m0) * sc); sum0 += e0[j];
      e1[j] = __expf((bufA[j * D + f1] - m1) * sc); sum1 += e1[j];
    }
    const float r0 = 1.f / sum0, r1 = 1.f / sum1;

    // res = sum_j attn * (v + pos_enc); emit attn
    float acc0 = 0.f, acc1 = 0.f;
    for (int j = 0; j < KNN; ++j) {
      const size_t nb = (size_t)b * NPTS + idxs[j];
      const float a0 = e0[j] * r0, a1 = e1[j] * r1;
      out_attn[((size_t)p * KNN + j) * D + f0] = a0;
      out_attn[((size_t)p * KNN + j) * D + f1] = a1;
      acc0 += a0 * (vi[nb * D + f0] + pos[j * D + f0]);
      acc1 += a1 * (vi[nb * D + f1] + pos[j * D + f1]);
    }
    resr[f0] = acc0;
    resr[f1] = acc1;

    // out = res @ fc2 + fc2_b + pre(features)
    float o0 = s_fc2b[f0] + features[(size_t)p * D + f0];
    float o1 = s_fc2b[f1] + features[(size_t)p * D + f1];
    for (int c = 0; c < D; ++c) {
      const float rv = resr[c];
      o0 += rv * s_fc2[c * D + f0];
      o1 += rv * s_fc2[c * D + f1];
    }
    out_res[(size_t)p * D + f0] = o0;
    out_res[(size_t)p * D + f1] = o1;
  }
}

extern "C" void kernel_launch(void* const* d_in, const int* in_sizes, int n_in,
                              void* d_out, int out_size, void* d_ws,
                              size_t ws_size, hipStream_t stream) {
  (void)in_sizes; (void)n_in; (void)out_size; (void)ws_size;
  const float* xyz      = (const float*)d_in[0];
  const float* features = (const float*)d_in[1];
  const int*   knn      = (const int*)d_in[2];
  const float* fc1_w = (const float*)d_in[3];
  const float* fc1_b = (const float*)d_in[4];
  const float* fc2_w = (const float*)d_in[5];
  const float* fc2_b = (const float*)d_in[6];
  const float* d1_w  = (const float*)d_in[7];
  const float* d1_b  = (const float*)d_in[8];
  const float* d2_w  = (const float*)d_in[9];
  const float* d2_b  = (const float*)d_in[10];
  const float* g1_w  = (const float*)d_in[11];
  const float* g1_b  = (const float*)d_in[12];
  const float* g2_w  = (const float*)d_in[13];
  const float* g2_b  = (const float*)d_in[14];
  const float* wq    = (const float*)d_in[15];
  const float* wk    = (const float*)d_in[16];
  const float* wv    = (const float*)d_in[17];

  float* qw = (float*)d_ws;
  float* kw = qw + (size_t)TOTAL_PTS * D;
  float* vw = kw + (size_t)TOTAL_PTS * D;
  float* out_res  = (float*)d_out;
  float* out_attn = out_res + (size_t)TOTAL_PTS * D;

  const size_t smem1 = (size_t)(4 * 4096 + 64 + 64 + 8 * 1024) * sizeof(float);
  const size_t smem2 = (size_t)(4 * 4096 + 192 + 5 * 64 + 4 * 2192) * sizeof(float);
  (void)hipFuncSetAttribute((const void*)proj_kernel,
                            hipFuncAttributeMaxDynamicSharedMemorySize, (int)smem1);
  (void)hipFuncSetAttribute((const void*)point_attn_kernel,
                            hipFuncAttributeMaxDynamicSharedMemorySize, (int)smem2);

  proj_kernel<<<512, 128, smem1, stream>>>(features, fc1_w, fc1_b, wq, wk, wv,
                                           qw, kw, vw);
  point_attn_kernel<<<1024, 128, smem2, stream>>>(
      xyz, features, knn, d1_w, d1_b, d2_w, d2_b, g1_w, g1_b, g2_w, g2_b,
      fc2_w, fc2_b, qw, kw, vw, out_res, out_attn);
}